// HbvAutograd_18854906429628
// MI455X (gfx1250) — compile-verified
//
#include <hip/hip_runtime.h>
#include <hip/hip_bf16.h>

#define T_STEPS 365
#define BGRID   20000
#define NMUL    4
#define B4      (BGRID * NMUL)
#define NEARZERO 1e-6f
#define NC      64            // cells per block (block = NC*NMUL = 256 threads)
#define NC3     (NC * 3)      // floats of forcing per timestep per block
#define TCHUNK  32            // timesteps staged per TDM transfer

typedef unsigned int u32x4 __attribute__((ext_vector_type(4)));
typedef int          i32x4 __attribute__((ext_vector_type(4)));
typedef int          i32x8 __attribute__((ext_vector_type(8)));

// Raw LDS byte offset of a __shared__ object (generic -> addrspace(3) -> int).
__device__ __forceinline__ unsigned lds_byte_offset(void* p) {
    return (unsigned)(unsigned long long)(__attribute__((address_space(3))) char*)p;
}

// Issue one TDM 2D tile load: tile_d0 contiguous elements per row, tile_d1 rows,
// rows separated by row_stride_elems elements, 4-byte elements, into LDS.
__device__ __forceinline__ void tdm_load_tile_2d(
    unsigned lds_off, unsigned long long gaddr,
    unsigned tensor_d0, unsigned tensor_d1,
    unsigned tile_d0, unsigned tile_d1,
    unsigned long long row_stride_elems)
{
    u32x4 g0;
    g0.x = 1u;                                              // count=1 (valid user D#)
    g0.y = lds_off;                                         // lds_addr (bytes)
    g0.z = (unsigned)(gaddr & 0xffffffffull);               // global_addr[31:0]
    g0.w = (unsigned)((gaddr >> 32) & 0x01ffffffull)        // global_addr[56:32]
         | 0x80000000u;                                     // type=2 ("image")

    i32x8 g1;
    g1[0] = 0x20000;                                        // data_size=2 (4 bytes)
    g1[1] = (int)((tensor_d0 & 0xffffu) << 16);             // tensor_dim0[15:0]
    g1[2] = (int)((tensor_d0 >> 16) | ((tensor_d1 & 0xffffu) << 16));
    g1[3] = (int)((tensor_d1 >> 16) | ((tile_d0 & 0xffffu) << 16));
    g1[4] = (int)(tile_d1 & 0xffffu);                       // tile_dim1, tile_dim2=0
    g1[5] = (int)(row_stride_elems & 0xffffffffull);        // tensor_dim0_stride lo
    g1[6] = (int)((row_stride_elems >> 32) & 0xffffull);    // stride hi, dim1_stride=0
    g1[7] = 0;

    i32x4 z4 = {0, 0, 0, 0};
#if defined(__clang_major__) && (__clang_major__ >= 23)
    i32x8 z8 = {0, 0, 0, 0, 0, 0, 0, 0};
    __builtin_amdgcn_tensor_load_to_lds(g0, g1, z4, z4, z8, 0);
#else
    __builtin_amdgcn_tensor_load_to_lds(g0, g1, z4, z4, 0);
#endif
}

__global__ __launch_bounds__(256)
void hbv_tdm_kernel(const float* __restrict__ forcing,
                    const float* __restrict__ pBETA,  const float* __restrict__ pFC,
                    const float* __restrict__ pK0,    const float* __restrict__ pK1,
                    const float* __restrict__ pK2,    const float* __restrict__ pLP,
                    const float* __restrict__ pPERC,  const float* __restrict__ pUZL,
                    const float* __restrict__ pTT,    const float* __restrict__ pCFMAX,
                    const float* __restrict__ pCFR,   const float* __restrict__ pCWH,
                    const float* __restrict__ pBETAET,const float* __restrict__ pC,
                    float* __restrict__ qavg,         float* __restrict__ qsim)
{
    __shared__ float sbuf[2][TCHUNK * NC3];   // 2 x 24 KB double buffer

    const int ltid = threadIdx.x;
    const int b0   = blockIdx.x * NC;                 // first cell of this block
    const int g    = blockIdx.x * (NC * NMUL) + ltid; // == b*4 + m (param index)
    const int b    = g >> 2;
    const int cl   = ltid >> 2;                       // cell-in-block
    const bool active = (b < BGRID);

    // ---- parameters (coalesced, read-once -> non-temporal) ----
    float BETA = 1.f, FC = 100.f, K0 = 0.1f, K1 = 0.1f, K2 = 0.01f, LP = 0.5f,
          PERC = 1.f, UZL = 1.f, TT = 0.f, CFMAX = 1.f, CFR = 0.f, CWH = 0.f,
          BETAET = 1.f, C = 0.f;
    if (active) {
        BETA   = __builtin_nontemporal_load(pBETA + g);
        FC     = __builtin_nontemporal_load(pFC + g);
        K0     = __builtin_nontemporal_load(pK0 + g);
        K1     = __builtin_nontemporal_load(pK1 + g);
        K2     = __builtin_nontemporal_load(pK2 + g);
        LP     = __builtin_nontemporal_load(pLP + g);
        PERC   = __builtin_nontemporal_load(pPERC + g);
        UZL    = __builtin_nontemporal_load(pUZL + g);
        TT     = __builtin_nontemporal_load(pTT + g);
        CFMAX  = __builtin_nontemporal_load(pCFMAX + g);
        CFR    = __builtin_nontemporal_load(pCFR + g);
        CWH    = __builtin_nontemporal_load(pCWH + g);
        BETAET = __builtin_nontemporal_load(pBETAET + g);
        C      = __builtin_nontemporal_load(pC + g);
    }
    const float invFC   = 1.0f / FC;
    const float invLPFC = 1.0f / (LP * FC);
    const float CFRCFM  = CFR * CFMAX;

    // ---- state ----
    float snow = NEARZERO, melt = 0.f, sm = 0.f, suz = 0.f, slz = 0.f;

    const unsigned lds_base = lds_byte_offset(&sbuf[0][0]);
    const unsigned td0 = 3u * (unsigned)min(NC, BGRID - b0); // valid elems/row (OOB->0)
    const int nchunks = (T_STEPS + TCHUNK - 1) / TCHUNK;

    // prologue: stage chunk 0 into buffer 0 (one TDM issue; EXEC ignored by TDM)
    if (ltid < 32) {
        unsigned long long ga = (unsigned long long)(uintptr_t)forcing
                              + 4ull * 3ull * (unsigned long long)b0;
        int rows0 = min(TCHUNK, T_STEPS);
        tdm_load_tile_2d(lds_base, ga, td0, (unsigned)rows0, NC3, (unsigned)rows0,
                         3ull * (unsigned long long)BGRID);
    }

    for (int c = 0; c < nchunks; ++c) {
        const int t0   = c * TCHUNK;
        const int rows = min(TCHUNK, T_STEPS - t0);

        if (ltid < 32) {
            if (c + 1 < nchunks) {
                // prefetch next chunk into the other buffer, then wait for current
                const int t0n   = (c + 1) * TCHUNK;
                const int rowsn = min(TCHUNK, T_STEPS - t0n);
                unsigned long long ga = (unsigned long long)(uintptr_t)forcing
                    + 4ull * 3ull * ((unsigned long long)t0n * BGRID
                                     + (unsigned long long)b0);
                tdm_load_tile_2d(lds_base + ((unsigned)((c + 1) & 1)) * (TCHUNK * NC3 * 4u),
                                 ga, td0, (unsigned)rowsn, NC3, (unsigned)rowsn,
                                 3ull * (unsigned long long)BGRID);
                __builtin_amdgcn_s_wait_tensorcnt(1);  // current chunk complete
            } else {
                __builtin_amdgcn_s_wait_tensorcnt(0);
            }
        }
        __syncthreads();   // staged forcing visible to all waves

        const float* buf = &sbuf[c & 1][0];
        for (int r = 0; r < rows; ++r) {
            const int t = t0 + r;
            const float* f = buf + r * NC3 + cl * 3;
            const float p   = f[0];
            const float tv  = f[1];
            const float pet = f[2];

            // ---- HBV step (mirrors reference) ----
            const float temp_diff  = tv - TT;
            const bool  is_rain    = temp_diff > 0.0f;
            const float rain       = is_rain ? p : 0.0f;
            const float snow_in    = is_rain ? 0.0f : p;
            const float snow1      = snow + snow_in;
            const float pot_melt   = CFMAX * fmaxf(temp_diff, 0.0f);
            const float melt_amt   = fminf(pot_melt, snow1);
            const float snow2      = snow1 - melt_amt;
            const float melt1      = melt + melt_amt;
            const float pot_refr   = CFRCFM * fmaxf(-temp_diff, 0.0f);
            const float refreeze   = fminf(pot_refr, melt1);
            snow = snow2 + refreeze;
            const float melt2      = melt1 - refreeze;
            const float tosoil     = fmaxf(melt2 - CWH * snow, 0.0f);
            melt = melt2 - tosoil;

            const float swb        = fmaxf(sm * invFC, NEARZERO);
            const float soil_wet   = fminf(fmaxf(__powf(swb, BETA), 0.0f), 1.0f);
            const float recharge   = (rain + tosoil) * soil_wet;
            const float sm1        = sm + rain + tosoil - recharge;
            const float excess     = fmaxf(sm1 - FC, 0.0f);
            const float sm2        = sm1 - excess;
            const float ef1        = fminf(fmaxf(sm2 * invLPFC, 0.0f), 1.0f);
            const float evapf      = fminf(fmaxf(__powf(fmaxf(ef1, NEARZERO), BETAET), 0.0f), 1.0f);
            const float etact      = fminf(pet * evapf, sm2);
            const float sm_ae      = fmaxf(sm2 - etact, NEARZERO);
            const float sm_ratio   = fminf(sm_ae * invFC, 1.0f);
            const float capillary  = fminf(slz, C * slz * (1.0f - sm_ratio));
            sm = fmaxf(sm_ae + capillary, NEARZERO);
            const float slz_ac     = fmaxf(slz - capillary, NEARZERO);

            const float suz1       = suz + recharge + excess;
            const float perc       = fminf(suz1, PERC);
            const float suz2       = suz1 - perc;
            const float slz1       = slz_ac + perc;
            const float q0         = K0 * fmaxf(suz2 - UZL, 0.0f);
            const float suz3       = suz2 - q0;
            const float q1         = K1 * suz3;
            suz = suz3 - q1;
            const float q2         = K2 * slz1;
            slz = slz1 - q2;
            const float qq         = q0 + q1 + q2;

            // streaming, write-once outputs -> non-temporal stores
            if (active)
                __builtin_nontemporal_store(qq, qsim + (size_t)t * B4 + (size_t)g);

            // mean over nmul: quad (lanes 4k..4k+3) reduction, wave32
            float qs = qq + __shfl_xor(qq, 1, 32);
            qs      += __shfl_xor(qs, 2, 32);
            if (active && ((ltid & 3) == 0))
                __builtin_nontemporal_store(0.25f * qs,
                                            qavg + (size_t)t * BGRID + (size_t)b);
        }
        __syncthreads();   // all waves done reading buf before it is re-staged
    }
}

extern "C" void kernel_launch(void* const* d_in, const int* in_sizes, int n_in,
                              void* d_out, int out_size, void* d_ws, size_t ws_size,
                              hipStream_t stream) {
    (void)in_sizes; (void)n_in; (void)d_ws; (void)ws_size; (void)out_size;

    // setup_inputs() order: forcing, then _BOUNDS order:
    // parBETA, parFC, parK0, parK1, parK2, parLP, parPERC, parUZL,
    // parTT, parCFMAX, parCFR, parCWH, parBETAET, parC
    const float* forcing = (const float*)d_in[0];
    const float* pBETA   = (const float*)d_in[1];
    const float* pFC     = (const float*)d_in[2];
    const float* pK0     = (const float*)d_in[3];
    const float* pK1     = (const float*)d_in[4];
    const float* pK2     = (const float*)d_in[5];
    const float* pLP     = (const float*)d_in[6];
    const float* pPERC   = (const float*)d_in[7];
    const float* pUZL    = (const float*)d_in[8];
    const float* pTT     = (const float*)d_in[9];
    const float* pCFMAX  = (const float*)d_in[10];
    const float* pCFR    = (const float*)d_in[11];
    const float* pCWH    = (const float*)d_in[12];
    const float* pBETAET = (const float*)d_in[13];
    const float* pC      = (const float*)d_in[14];

    float* out  = (float*)d_out;
    float* qavg = out;                                   // (T, B)
    float* qsim = out + (size_t)T_STEPS * BGRID;         // (T, B, NMUL)

    const int blocks = (B4 + 255) / 256;                 // 313
    hipLaunchKernelGGL(hbv_tdm_kernel, dim3(blocks), dim3(256), 0, stream,
                       forcing, pBETA, pFC, pK0, pK1, pK2, pLP, pPERC, pUZL,
                       pTT, pCFMAX, pCFR, pCWH, pBETAET, pC, qavg, qsim);
}